// LinearVectorTransform_48515950575975
// MI455X (gfx1250) — compile-verified
//
#include <hip/hip_runtime.h>

// ---- problem constants (match reference) ----
#define BATCH 1024
#define VDIM  768
#define NACT  8

#define LDSSTRIDE (VDIM + 4)   // +4 floats pad -> conflict-free ds_load_b64

typedef __attribute__((ext_vector_type(2))) float v2f;
typedef __attribute__((ext_vector_type(8))) float v8f;

#ifndef __has_builtin
#define __has_builtin(x) 0
#endif

#if __has_builtin(__builtin_amdgcn_global_load_async_to_lds_b128)
#define USE_ASYNC_LDS 1
#else
#define USE_ASYNC_LDS 0
#endif

__device__ __forceinline__ void wait_async_zero() {
#if __has_builtin(__builtin_amdgcn_s_wait_asynccnt)
    __builtin_amdgcn_s_wait_asynccnt(0);
#else
    asm volatile("s_wait_asynccnt 0" ::: "memory");
#endif
}

// ---------------- binning: group batch rows by action ----------------
__global__ void lvt_zero_counts(int* __restrict__ counts) {
    if (threadIdx.x < NACT) counts[threadIdx.x] = 0;
}

__global__ void lvt_bin(const int* __restrict__ action,
                        int* __restrict__ counts,
                        int* __restrict__ rowlist) {
    int b = blockIdx.x * blockDim.x + threadIdx.x;
    if (b < BATCH) {
        int a = action[b];
        int pos = atomicAdd(&counts[a], 1);
        rowlist[a * BATCH + pos] = b;
    }
}

// ------- grouped GEMM: 32x128 tile per block, WMMA f32 16x16x4 -------
// grid = (VDIM/128, BATCH/32, NACT), block = 256 (8 waves).
// Each wave owns one 16-col N-subtile and BOTH 16-row M-subtiles, so each
// B fragment feeds two WMMAs (independent accumulator chains).
__global__ __launch_bounds__(256) void lvt_gemm(
    const float* __restrict__ before,   // [BATCH, VDIM]
    const float* __restrict__ weights,  // [NACT, VDIM, VDIM]
    const int*   __restrict__ counts,   // [NACT]
    const int*   __restrict__ rowlist,  // [NACT, BATCH]
    float* __restrict__ out) {          // [BATCH, VDIM]

    const int a   = blockIdx.z;
    const int cnt = counts[a];
    const int rt  = blockIdx.y;                 // 32-row tile within action group
    if (rt * 32 >= cnt) return;
    const int ct  = blockIdx.x;                 // 128-col tile

    __shared__ int   rows[32];
    __shared__ float As[32][LDSSTRIDE];

    const int tid = threadIdx.x;

    if (tid < 32) {
        int rg = rt * 32 + tid;
        rows[tid] = (rg < cnt) ? rowlist[a * BATCH + rg] : -1;
    }
    __syncthreads();

    // Stage A tile (gathered rows) into LDS: 32 rows x 768 f32.
#if USE_ASYNC_LDS
    typedef int v4i __attribute__((vector_size(16)));
    typedef __attribute__((address_space(1))) v4i as1_v4i;
    typedef __attribute__((address_space(3))) v4i as3_v4i;
    for (int idx = tid; idx < 32 * (VDIM / 4); idx += 256) {
        int m   = idx / (VDIM / 4);
        int c4  = idx % (VDIM / 4);
        int row = rows[m];
        float* lp = &As[m][c4 * 4];
        if (row >= 0) {
            const float* gp = before + (size_t)row * VDIM + (size_t)c4 * 4;
            __builtin_amdgcn_global_load_async_to_lds_b128(
                (as1_v4i*)(size_t)gp,
                (as3_v4i*)(unsigned)(size_t)lp,
                0, 0);
        } else {
            *(float4*)lp = make_float4(0.f, 0.f, 0.f, 0.f);
        }
    }
    wait_async_zero();
    __syncthreads();
#else
    for (int idx = tid; idx < 32 * (VDIM / 4); idx += 256) {
        int m   = idx / (VDIM / 4);
        int c4  = idx % (VDIM / 4);
        int row = rows[m];
        float4 v = make_float4(0.f, 0.f, 0.f, 0.f);
        if (row >= 0)
            v = *(const float4*)(before + (size_t)row * VDIM + (size_t)c4 * 4);
        *(float4*)&As[m][c4 * 4] = v;
    }
    __syncthreads();
#endif

    const int wave  = tid >> 5;        // 0..7 -> N subtile
    const int lane  = tid & 31;
    const int khalf = lane >> 4;       // K-half per ISA f32 A/B layout
    const int m_a   = lane & 15;       // A row (within subtile) for this lane
    const int n0    = ct * 128 + wave * 16 + (lane & 15);  // output column

    // B: per lane, rows (k+2*khalf) and (k+2*khalf+1) at column n0
    const float* __restrict__ wp =
        weights + (size_t)a * VDIM * VDIM + (size_t)(2 * khalf) * VDIM + n0;
    // A: contiguous float2 at [m][k + 2*khalf] for both M subtiles
    const float* __restrict__ ap0 = &As[m_a][2 * khalf];
    const float* __restrict__ ap1 = &As[m_a + 16][2 * khalf];

    v8f acc0 = {0.f, 0.f, 0.f, 0.f, 0.f, 0.f, 0.f, 0.f};
    v8f acc1 = {0.f, 0.f, 0.f, 0.f, 0.f, 0.f, 0.f, 0.f};

    #pragma unroll 4
    for (int k0 = 0; k0 < VDIM; k0 += 4) {
        v2f bv;
        bv.x = wp[0];
        bv.y = wp[VDIM];
        v2f a0 = *(const v2f*)ap0;
        v2f a1 = *(const v2f*)ap1;
        acc0 = __builtin_amdgcn_wmma_f32_16x16x4_f32(
            false, a0, false, bv, (short)0, acc0, false, false);
        acc1 = __builtin_amdgcn_wmma_f32_16x16x4_f32(
            false, a1, false, bv, (short)0, acc1, false, false);
        ap0 += 4;
        ap1 += 4;
        wp  += 4 * VDIM;
    }

    // D layout: VGPR r -> M = r + 8*khalf (per subtile), N = n0
    const int mbase = 8 * khalf;
    #pragma unroll
    for (int r = 0; r < 8; ++r) {
        int row0 = rows[mbase + r];
        if (row0 >= 0)
            out[(size_t)row0 * VDIM + n0] = acc0[r];
        int row1 = rows[16 + mbase + r];
        if (row1 >= 0)
            out[(size_t)row1 * VDIM + n0] = acc1[r];
    }
}

extern "C" void kernel_launch(void* const* d_in, const int* in_sizes, int n_in,
                              void* d_out, int out_size, void* d_ws, size_t ws_size,
                              hipStream_t stream) {
    const float* before  = (const float*)d_in[0];   // [1024, 768] f32
    const int*   action  = (const int*)d_in[1];     // [1024] int
    const float* weights = (const float*)d_in[2];   // [8, 768, 768] f32
    float* out = (float*)d_out;                     // [1024, 768] f32

    // workspace layout: counts[8] | rowlist[8*1024]
    int* counts  = (int*)d_ws;
    int* rowlist = counts + NACT;

    lvt_zero_counts<<<1, 32, 0, stream>>>(counts);
    lvt_bin<<<(BATCH + 255) / 256, 256, 0, stream>>>(action, counts, rowlist);

    dim3 grid(VDIM / 128, BATCH / 32, NACT);
    lvt_gemm<<<grid, 256, 0, stream>>>(before, weights, counts, rowlist, out);
}